// TwoConvThreeClassiLayerGCN_50448685859135
// MI455X (gfx1250) — compile-verified
//
#include <hip/hip_runtime.h>

#define N_NODES   50000
#define N_EDGES   800000
#define N_GRAPHS  64
#define IN_DIM    128
#define HID1      256
#define HID2      128
#define N_CLASSES 10

typedef float v2f __attribute__((ext_vector_type(2)));
typedef float v8f __attribute__((ext_vector_type(8)));

// ---------------------------------------------------------------- degrees
__global__ void k_degree(const int* __restrict__ src, const int* __restrict__ dst,
                         float* __restrict__ deg_o, float* __restrict__ deg_i) {
  int e = blockIdx.x * blockDim.x + threadIdx.x;
  if (e < N_EDGES) {
    atomicAdd(&deg_o[src[e]], 1.0f);
    atomicAdd(&deg_i[dst[e]], 1.0f);
  }
}

__global__ void k_invdeg(const float* __restrict__ deg_o, const float* __restrict__ deg_i,
                         float* __restrict__ inv_o, float* __restrict__ inv_i) {
  int n = blockIdx.x * blockDim.x + threadIdx.x;
  if (n < N_NODES) {
    inv_o[n] = rsqrtf(fmaxf(deg_o[n], 1.0f));
    inv_i[n] = rsqrtf(fmaxf(deg_i[n], 1.0f));
  }
}

// ------------------------------------------------- weight transpose (prep)
// W[K][M] -> Wt[M][K]; tiny one-time cost, makes every WMMA B-fragment a
// single contiguous per-lane b64 load with an immediate offset.
__global__ void k_transpose(const float* __restrict__ W, float* __restrict__ Wt,
                            int K, int M) {
  int idx = blockIdx.x * blockDim.x + threadIdx.x;
  if (idx < K * M) {
    int k = idx / M, m = idx - k * M;
    Wt[m * K + k] = W[idx];
  }
}

// ------------------------------------------------- edge aggregation (SpMM)
// One wave per edge; 128 features = one float4 per lane. Source-feature
// gathers and destination atomics both hit L2 (feature arrays < 192MB L2).
template <bool SCALED>
__global__ void k_edge_agg(const float* __restrict__ feat, const float* __restrict__ scale,
                           const int* __restrict__ src, const int* __restrict__ dst,
                           float* __restrict__ out) {
  int wave = (blockIdx.x * blockDim.x + threadIdx.x) >> 5;
  int lane = threadIdx.x & 31;
  if (wave >= N_EDGES) return;
  int s = src[wave];
  int d = dst[wave];
  float sc = SCALED ? scale[s] : 1.0f;
  float4 v = ((const float4*)(feat + (size_t)s * 128))[lane];
  float* o = out + (size_t)d * 128 + lane * 4;
  atomicAdd(o + 0, v.x * sc);
  atomicAdd(o + 1, v.y * sc);
  atomicAdd(o + 2, v.z * sc);
  atomicAdd(o + 3, v.w * sc);
}

// ----------------------------------------------------- WMMA fp32 GEMM
// D[N,M] = op(A[N,K]) @ W[K,M] with W pre-transposed as Wt[M][K].
// One wave per 16x128 output tile: 8 f32 accumulators. Per k-step: one A
// b64 + eight B b64 loads (batched, fresh registers -> one s_clause + one
// wait), then 8 v_wmma_f32_16x16x4_f32. Both streams use a single VGPR base
// with compile-time immediate offsets (K is a template constant), and the k
// loop is fully unrolled, so the hot code has zero address arithmetic.
// A fragment (ISA 7.12.2): lanes 0-15 carry K=k,k+1; lanes 16-31 K=k+2,k+3
// for row = row0 + (lane&15). B fragment mirrored via the transpose.
template <int K, bool PRESCALE, bool EPILOGUE>
__global__ void k_gemm_wmma(const float* __restrict__ A, const float* __restrict__ Wt,
                            const float* __restrict__ preScale,
                            const float* __restrict__ postScale,
                            const float* __restrict__ bias,
                            float* __restrict__ D, int M) {
  const int lane = threadIdx.x & 31;
  const int row0 = blockIdx.x * 16;
  const int col0 = blockIdx.y * 128;
  const int half = lane >> 4;
  const int lr   = lane & 15;
  const int r    = row0 + lr;
  const float rs = PRESCALE ? preScale[r] : 1.0f;

  const float* __restrict__ abase = A + (size_t)r * K + 2 * half;
  const float* __restrict__ bbase = Wt + (size_t)(col0 + lr) * K + 2 * half;

  v8f c[8] = {};
#pragma unroll
  for (int k = 0; k < K; k += 4) {
    v2f a = *(const v2f*)(abase + k);
    a.x *= rs;
    a.y *= rs;
    v2f b[8];
#pragma unroll
    for (int t = 0; t < 8; ++t)
      b[t] = *(const v2f*)(bbase + 16 * t * K + k);   // immediate offsets
#pragma unroll
    for (int t = 0; t < 8; ++t)
      c[t] = __builtin_amdgcn_wmma_f32_16x16x4_f32(false, a, false, b[t], (short)0,
                                                   c[t], false, false);
  }

  // C/D layout: VGPR v holds row = row0 + v + 8*half, col = col0 + lr + 16*t
  const int erow = row0 + 8 * half;
#pragma unroll
  for (int v = 0; v < 8; ++v) {
    const int rr = erow + v;
    float* __restrict__ Dp = D + (size_t)rr * M + col0 + lr;
    if (EPILOGUE) {
      const float p = postScale[rr];
#pragma unroll
      for (int t = 0; t < 8; ++t)
        Dp[16 * t] = fmaxf(fmaf(c[t][v], p, bias[col0 + lr + 16 * t]), 0.0f);
    } else {
#pragma unroll
      for (int t = 0; t < 8; ++t)
        Dp[16 * t] = c[t][v];
    }
  }
}

// ---------------- fused layer-2 epilogue (relu(agg*inv_i+b2)) + mean-pool
__global__ void k_pool(const float* __restrict__ agg2, const float* __restrict__ inv_i,
                       const float* __restrict__ b2, const int* __restrict__ gid,
                       float* __restrict__ gsum, float* __restrict__ cnt) {
  int wave = (blockIdx.x * blockDim.x + threadIdx.x) >> 5;
  int lane = threadIdx.x & 31;
  if (wave >= N_NODES) return;
  int g = gid[wave];
  float si = inv_i[wave];
  float4 v = ((const float4*)(agg2 + (size_t)wave * HID2))[lane];
  float4 b = ((const float4*)b2)[lane];
  float* o = gsum + g * HID2 + lane * 4;
  atomicAdd(o + 0, fmaxf(fmaf(v.x, si, b.x), 0.0f));
  atomicAdd(o + 1, fmaxf(fmaf(v.y, si, b.y), 0.0f));
  atomicAdd(o + 2, fmaxf(fmaf(v.z, si, b.z), 0.0f));
  atomicAdd(o + 3, fmaxf(fmaf(v.w, si, b.w), 0.0f));
  if (lane == 0) atomicAdd(&cnt[g], 1.0f);
}

// --------------------------------------------- tiny 3-layer classifier MLP
__global__ void k_mlp(const float* __restrict__ gsum, const float* __restrict__ cnt,
                      const float* __restrict__ Wc1, const float* __restrict__ bc1,
                      const float* __restrict__ Wc2, const float* __restrict__ bc2,
                      const float* __restrict__ Wc3, const float* __restrict__ bc3,
                      float* __restrict__ out) {
  int g = threadIdx.x;
  if (g >= N_GRAPHS) return;
  float inv = 1.0f / fmaxf(cnt[g], 1.0f);
  const float* __restrict__ hg = gsum + g * HID2;
  float t1[12];
#pragma unroll
  for (int j = 0; j < 12; ++j) {
    float s = 0.0f;
    for (int i = 0; i < HID2; ++i) s = fmaf(hg[i], Wc1[i * 12 + j], s);
    t1[j] = fmaf(s, inv, bc1[j]);   // mean applied after linear combine
  }
  float t2[12];
#pragma unroll
  for (int j = 0; j < 12; ++j) {
    float s = bc2[j];
#pragma unroll
    for (int i = 0; i < 12; ++i) s = fmaf(t1[i], Wc2[i * 12 + j], s);
    t2[j] = s;
  }
#pragma unroll
  for (int k = 0; k < N_CLASSES; ++k) {
    float s = bc3[k];
#pragma unroll
    for (int j = 0; j < 12; ++j) s = fmaf(t2[j], Wc3[j * N_CLASSES + k], s);
    out[g * N_CLASSES + k] = s;
  }
}

extern "C" void kernel_launch(void* const* d_in, const int* in_sizes, int n_in,
                              void* d_out, int out_size, void* d_ws, size_t ws_size,
                              hipStream_t stream) {
  const float* x   = (const float*)d_in[0];
  const int*   src = (const int*)  d_in[1];
  const int*   dst = (const int*)  d_in[2];
  const int*   gid = (const int*)  d_in[3];
  const float* W1  = (const float*)d_in[4];
  const float* b1  = (const float*)d_in[5];
  const float* W2  = (const float*)d_in[6];
  const float* b2  = (const float*)d_in[7];
  const float* Wc1 = (const float*)d_in[8];
  const float* bc1 = (const float*)d_in[9];
  const float* Wc2 = (const float*)d_in[10];
  const float* bc2 = (const float*)d_in[11];
  const float* Wc3 = (const float*)d_in[12];
  const float* bc3 = (const float*)d_in[13];
  float* out = (float*)d_out;

  // workspace carve-up (fp32): ~104 MB total
  float* ws    = (float*)d_ws;
  float* deg_o = ws;                                   // 50000
  float* deg_i = deg_o + N_NODES;                      // 50000
  float* inv_o = deg_i + N_NODES;                      // 50000
  float* inv_i = inv_o + N_NODES;                      // 50000
  float* aggX  = inv_i + N_NODES;                      // 50000*128 (reused as xw2)
  float* h1    = aggX + (size_t)N_NODES * IN_DIM;      // 50000*256
  float* agg2  = h1   + (size_t)N_NODES * HID1;        // 50000*128
  float* gsum  = agg2 + (size_t)N_NODES * HID2;        // 64*128
  float* cnt   = gsum + N_GRAPHS * HID2;               // 64
  float* Wt1   = cnt  + N_GRAPHS;                      // 256*128 (W1^T)
  float* Wt2   = Wt1  + HID1 * IN_DIM;                 // 128*256 (W2^T)

  hipMemsetAsync(deg_o, 0, (size_t)2 * N_NODES * sizeof(float), stream);
  hipMemsetAsync(aggX,  0, (size_t)N_NODES * IN_DIM * sizeof(float), stream);
  hipMemsetAsync(agg2,  0, (size_t)N_NODES * HID2 * sizeof(float), stream);
  hipMemsetAsync(gsum,  0, (size_t)(N_GRAPHS * HID2 + N_GRAPHS) * sizeof(float), stream);

  k_degree<<<(N_EDGES + 255) / 256, 256, 0, stream>>>(src, dst, deg_o, deg_i);
  k_invdeg<<<(N_NODES + 255) / 256, 256, 0, stream>>>(deg_o, deg_i, inv_o, inv_i);
  k_transpose<<<(IN_DIM * HID1 + 255) / 256, 256, 0, stream>>>(W1, Wt1, IN_DIM, HID1);
  k_transpose<<<(HID1 * HID2 + 255) / 256, 256, 0, stream>>>(W2, Wt2, HID1, HID2);

  // Layer 1: aggregate scaled 128-dim inputs first (linearity), then GEMM.
  k_edge_agg<true><<<N_EDGES / 8, 256, 0, stream>>>(x, inv_o, src, dst, aggX);
  // h1 = relu((aggX @ W1) * inv_i + b1)            [50000,128]@[128,256]
  k_gemm_wmma<IN_DIM, false, true>
      <<<dim3(N_NODES / 16, HID1 / 128), 32, 0, stream>>>(aggX, Wt1, nullptr, inv_i, b1, h1, HID1);
  // Layer 2: xw2 = (h1 * inv_o) @ W2 (into aggX)   [50000,256]@[256,128]
  k_gemm_wmma<HID1, true, false>
      <<<dim3(N_NODES / 16, HID2 / 128), 32, 0, stream>>>(h1, Wt2, inv_o, nullptr, nullptr, aggX, HID2);
  k_edge_agg<false><<<N_EDGES / 8, 256, 0, stream>>>(aggX, nullptr, src, dst, agg2);

  // fused relu(agg2*inv_i + b2) + per-graph sum/count, then tiny MLP
  k_pool<<<N_NODES / 8, 256, 0, stream>>>(agg2, inv_i, b2, gid, gsum, cnt);
  k_mlp<<<1, 64, 0, stream>>>(gsum, cnt, Wc1, bc1, Wc2, bc2, Wc3, bc3, out);
}